// ChebTimeConv_13288628814254
// MI455X (gfx1250) — compile-verified
//
#include <hip/hip_runtime.h>

#define N_NODES 50000
#define E_EDGES 1600000
#define HF      128     // H * F_IN
#define FOUT    32
#define KTOT    512     // KCHEB * HF

typedef __attribute__((ext_vector_type(2))) float v2f;
typedef __attribute__((ext_vector_type(8))) float v8f;

// ---------------- zero (float4 grid-stride) ----------------
__global__ void zero4_kernel(float4* __restrict__ p, long n4) {
    long i = (long)blockIdx.x * blockDim.x + threadIdx.x;
    long stride = (long)gridDim.x * blockDim.x;
    float4 z = {0.f, 0.f, 0.f, 0.f};
    for (; i < n4; i += stride) p[i] = z;
}

// ---------------- degree ----------------
__global__ void degree_kernel(const long long* __restrict__ ei, float* __restrict__ deg) {
    int e = blockIdx.x * blockDim.x + threadIdx.x;
    if (e >= E_EDGES) return;
    long long r = ei[e];
    long long c = ei[E_EDGES + e];
    if (r != c) atomicAdd(&deg[r], 1.0f);
}

// ---------------- dinv (in place over deg) ----------------
__global__ void dinv_kernel(float* __restrict__ deg) {
    int n = blockIdx.x * blockDim.x + threadIdx.x;
    if (n >= N_NODES) return;
    float d = deg[n];
    deg[n] = (d > 0.f) ? rsqrtf(d) : 0.f;
}

// ---------------- per-edge laplacian weight ----------------
__global__ void lap_kernel(const long long* __restrict__ ei,
                           const float* __restrict__ ew,
                           const float* __restrict__ dinv,
                           float* __restrict__ lap) {
    int e = blockIdx.x * blockDim.x + threadIdx.x;
    if (e >= E_EDGES) return;
    long long r = ei[e];
    long long c = ei[E_EDGES + e];
    lap[e] = (r != c) ? (-dinv[r] * ew[e] * dinv[c]) : 0.f;
}

// ---------------- SpMM: dst[row] += lap * src[col], one wave per edge ----------------
__global__ void spmm_kernel(const long long* __restrict__ ei,
                            const float* __restrict__ lap,
                            const float* __restrict__ src,
                            float* __restrict__ dst) {
    int gtid = blockIdx.x * blockDim.x + threadIdx.x;
    int e = gtid >> 5;           // one wave32 per edge
    int lane = gtid & 31;
    if (e >= E_EDGES) return;
    float l = lap[e];
    if (l == 0.f) return;        // masked self-loop: no work
    long long r = ei[e];
    long long c = ei[E_EDGES + e];
    const float4 v = *(const float4*)(src + (size_t)c * HF + lane * 4);
    float* d = dst + (size_t)r * HF + lane * 4;
    atomicAdd(d + 0, l * v.x);
    atomicAdd(d + 1, l * v.y);
    atomicAdd(d + 2, l * v.z);
    atomicAdd(d + 3, l * v.w);
}

// ---------------- Chebyshev combine: S = 2*S - P (in place) ----------------
__global__ void combine_kernel(float4* __restrict__ S, const float4* __restrict__ P) {
    long i = (long)blockIdx.x * blockDim.x + threadIdx.x;
    if (i >= (long)N_NODES * HF / 4) return;
    float4 s = S[i];
    float4 p = P[i];
    s.x = 2.f * s.x - p.x;
    s.y = 2.f * s.y - p.y;
    s.z = 2.f * s.z - p.z;
    s.w = 2.f * s.w - p.w;
    S[i] = s;
}

// ---------------- WMMA GEMM: out(50000x32) = [Tx0|Tx1|Tx2|Tx3](50000x512) @ W(512x32) + bias ----------------
// V_WMMA_F32_16X16X4_F32 fragment layouts (wave32):
//   A 16x4 : lane L holds row M=L&15; VGPR0/1 = K = kbase+2*(L>>4), +1
//   B 4x16 : lane L holds col N=L&15;          VGPR0/1 = K = kbase+2*(L>>4), +1
//   C/D    : v8f; VGPR r -> M = 8*(L>>4)+r, N = L&15
__device__ inline void accum128(const float* __restrict__ row,    // per-lane A row base (node*HF)
                                const float* __restrict__ wlds,   // LDS weight block (128 x 32)
                                int lane, v8f& c0, v8f& c1) {
    const int kq  = (lane >> 4) << 1;   // 0 or 2
    const int col = lane & 15;
#pragma unroll 4
    for (int kk = 0; kk < HF; kk += 4) {
        const int kl = kk + kq;
        v2f a = *(const v2f*)(row + kl);                 // 8B load, kl is even
        const float* w0 = wlds + (size_t)kl * FOUT;
        v2f b0 = { w0[col],      w0[FOUT + col]      };
        v2f b1 = { w0[16 + col], w0[FOUT + 16 + col] };
        c0 = __builtin_amdgcn_wmma_f32_16x16x4_f32(false, a, false, b0, (short)0, c0, false, false);
        c1 = __builtin_amdgcn_wmma_f32_16x16x4_f32(false, a, false, b1, (short)0, c1, false, false);
    }
}

__global__ __launch_bounds__(256) void cheb_gemm_wmma(
    const float* __restrict__ tx0, const float* __restrict__ tx1,
    const float* __restrict__ tx2, const float* __restrict__ tx3,
    const float* __restrict__ weight, const float* __restrict__ bias,
    float* __restrict__ out) {
    __shared__ float Wlds[KTOT * FOUT];   // 64 KB: entire 512x32 weight
    for (int i = threadIdx.x; i < KTOT * FOUT; i += blockDim.x)
        Wlds[i] = weight[i];
    __syncthreads();

    const int lane = threadIdx.x & 31;
    const int tile = blockIdx.x * 8 + (threadIdx.x >> 5);   // 16 nodes per wave-tile
    if (tile * 16 >= N_NODES) return;                       // wave-uniform (EXEC stays full)

    const size_t rowoff = (size_t)(tile * 16 + (lane & 15)) * HF;

    v8f c0 = {};
    v8f c1 = {};
    accum128(tx0 + rowoff, Wlds + 0 * HF * FOUT, lane, c0, c1);
    accum128(tx1 + rowoff, Wlds + 1 * HF * FOUT, lane, c0, c1);
    accum128(tx2 + rowoff, Wlds + 2 * HF * FOUT, lane, c0, c1);
    accum128(tx3 + rowoff, Wlds + 3 * HF * FOUT, lane, c0, c1);

    const int col = lane & 15;
    const float b0 = bias[col];
    const float b1 = bias[16 + col];
    const int mbase = (lane >> 4) * 8;
#pragma unroll
    for (int r = 0; r < 8; ++r) {
        const size_t n = (size_t)(tile * 16 + mbase + r) * FOUT;
        out[n + col]      = c0[r] + b0;
        out[n + 16 + col] = c1[r] + b1;
    }
}

extern "C" void kernel_launch(void* const* d_in, const int* in_sizes, int n_in,
                              void* d_out, int out_size, void* d_ws, size_t ws_size,
                              hipStream_t stream) {
    const float*     x    = (const float*)d_in[0];       // (N,H,F_IN,1) = N x 128
    const long long* ei   = (const long long*)d_in[1];   // (2,E) int64
    const float*     ew   = (const float*)d_in[2];       // (E,)
    const float*     wgt  = (const float*)d_in[3];       // (4,4,32,32) = 512 x 32 row-major
    const float*     bias = (const float*)d_in[4];       // (32,)
    float*           out  = (float*)d_out;               // N x 32

    // workspace layout
    float* deg = (float*)d_ws;                    // N   (becomes dinv in place)
    float* lap = deg + N_NODES;                   // E
    float* S1  = lap + E_EDGES;                   // N*HF  (Tx1)
    float* S2  = S1 + (size_t)N_NODES * HF;       // N*HF  (spmm(Tx1) -> Tx2)
    float* S3  = S2 + (size_t)N_NODES * HF;       // N*HF  (spmm(Tx2) -> Tx3)

    // deterministic re-init of atomic accumulators every call
    zero4_kernel<<<1024, 256, 0, stream>>>((float4*)deg, N_NODES / 4);
    zero4_kernel<<<4096, 256, 0, stream>>>((float4*)S1, (long)3 * N_NODES * HF / 4);

    const int EB = (E_EDGES + 255) / 256;
    degree_kernel<<<EB, 256, 0, stream>>>(ei, deg);
    dinv_kernel<<<(N_NODES + 255) / 256, 256, 0, stream>>>(deg);
    lap_kernel<<<EB, 256, 0, stream>>>(ei, ew, deg, lap);

    const int SPMMB = E_EDGES / 8;                // one wave32 per edge, 8 waves/block
    const int CMB   = (N_NODES * HF / 4 + 255) / 256;

    spmm_kernel<<<SPMMB, 256, 0, stream>>>(ei, lap, x, S1);     // Tx1 = L x
    spmm_kernel<<<SPMMB, 256, 0, stream>>>(ei, lap, S1, S2);
    combine_kernel<<<CMB, 256, 0, stream>>>((float4*)S2, (const float4*)x);   // Tx2 = 2 L Tx1 - Tx0
    spmm_kernel<<<SPMMB, 256, 0, stream>>>(ei, lap, S2, S3);
    combine_kernel<<<CMB, 256, 0, stream>>>((float4*)S3, (const float4*)S1);  // Tx3 = 2 L Tx2 - Tx1

    const int TILES = N_NODES / 16;               // 3125
    cheb_gemm_wmma<<<(TILES + 7) / 8, 256, 0, stream>>>(x, S1, S2, S3, wgt, bias, out);
}